// GuassianFilter_39144331936128
// MI455X (gfx1250) — compile-verified
//
#include <hip/hip_runtime.h>
#include <math.h>

// CDNA5 (gfx1250) fp32 WMMA types: 16x16x4_f32 -> A,B = 2 VGPRs (v2f), C/D = 8 VGPRs (v8f)
typedef __attribute__((ext_vector_type(2))) float v2f;
typedef __attribute__((ext_vector_type(8))) float v8f;
typedef __attribute__((ext_vector_type(4))) float f4;

#define H 64
#define W 64
#define IN_STRIDE 72   // padded LDS row stride (floats); left pad 4 keeps float4 stores 16B aligned
#define PAD_LEFT  4
#define TMP_ROWS  68   // 2 + 64 + 2 zero-halo rows for vertical pass
#define TMP_STRIDE 64

// Banded Gaussian weight: w(d) = g1d[d] for d in [0,4], else 0.
// g1d symmetric: g1d[0]=g1d[4]=g0, g1d[1]=g1d[3]=g1, g1d[2]=g2.
__device__ __forceinline__ float wband(int d, float g0, float g1, float g2) {
    float w = 0.0f;
    w = (d == 0 || d == 4) ? g0 : w;
    w = (d == 1 || d == 3) ? g1 : w;
    w = (d == 2)           ? g2 : w;
    return w;
}

__global__ __launch_bounds__(256) void gauss_sep_wmma(
    const float* __restrict__ x, const float* __restrict__ sigma,
    float* __restrict__ out)
{
    __shared__ __align__(16) float s_in [H * IN_STRIDE];         // input plane, col halo padded
    __shared__ __align__(16) float s_tmp[TMP_ROWS * TMP_STRIDE]; // horizontal result, row halo padded
    __shared__ __align__(8)  v2f   s_wb [5][32];                 // per-lane band-weight chunks

    const int tid   = threadIdx.x;
    const int plane = blockIdx.x;                 // n*C + c
    const float* __restrict__ src = x   + (size_t)plane * (H * W);
    float*       __restrict__ dst = out + (size_t)plane * (H * W);

    const int wv   = tid >> 5;                    // 0..7 (wave32)
    const int lane = tid & 31;
    const int l16  = lane & 15;
    const int hi   = lane >> 4;                   // 0: K={0,1}, 1: K={2,3} (A/B half-wave split)
    const int tx   = wv & 3;                      // tile col 0..3
    const int ty0  = wv >> 2;                     // tile rows: ty0 (0..1) and ty0+2 (2..3)
    const int ty1  = ty0 + 2;

    // ---- wave 0 ONLY: Gaussian taps + per-lane band-weight table into LDS ----
    // Per-lane weights depend only on lane, so one wave computes them for everyone.
    // Horizontal B[i][j]=g1d[i-j] and vertical A[y][i]=g1d[i-y] reduce (by the A vs B
    // VGPR layouts) to the SAME per-lane values: wband(kk*4 + 2*hi + {0,1} - l16).
    if (wv == 0) {
        const float s   = sigma[0];
        const float inv = 1.0f / (2.0f * s * s);
        const float e1  = expf(-1.0f * inv);
        const float e4  = expf(-4.0f * inv);
        const float nrm = 1.0f / (1.0f + 2.0f * (e1 + e4));
        const float g0 = e4 * nrm, g1 = e1 * nrm, g2 = nrm;  // normalized 1-D taps
        #pragma unroll
        for (int kk = 0; kk < 5; ++kk) {
            int i0 = kk * 4 + hi * 2;
            v2f w;
            w.x = wband(i0     - l16, g0, g1, g2);
            w.y = wband(i0 + 1 - l16, g0, g1, g2);
            s_wb[kk][lane] = w;
        }
    }

    // ---- zero column halos of s_in: cols 0..3 and 68..71 (512 floats, 2/thread) ----
    #pragma unroll
    for (int k = 0; k < 2; ++k) {
        int t = tid + k * 256;
        int r = t >> 3;
        int c = t & 7;
        int col = (c < 4) ? c : (64 + c);         // 0..3 , 68..71
        s_in[r * IN_STRIDE + col] = 0.0f;
    }
    // ---- zero row halos of s_tmp: rows 0,1,66,67 (256 floats, 1/thread) ----
    {
        int rr  = tid >> 6;                       // 0..3
        int row = (rr < 2) ? rr : (64 + rr);      // 0,1,66,67
        s_tmp[row * TMP_STRIDE + (tid & 63)] = 0.0f;
    }

    // ---- coalesced load of the 64x64 plane (1024 float4, 4 per thread) ----
    const f4* __restrict__ src4 = (const f4*)src;
    #pragma unroll
    for (int k = 0; k < 4; ++k) {
        int i   = tid + k * 256;                  // float4 index
        f4  v   = src4[i];
        int row = i >> 4;                         // 16 float4 per row
        int c4  = i & 15;
        *(f4*)&s_in[row * IN_STRIDE + PAD_LEFT + c4 * 4] = v;  // 16B aligned: 72r+4+4c
    }
    __syncthreads();

    // ---- fetch band-weight chunks (LDS broadcast-ish loads, no VALU chains) ----
    v2f wb[5];
    #pragma unroll
    for (int kk = 0; kk < 5; ++kk) wb[kk] = s_wb[kk][lane];

    // ========== horizontal pass: Tmp = In(16x20) x Band(20x16), 2 tiles/wave ==========
    // Interleave the two accumulator chains to break WMMA->WMMA RAW serialization.
    v8f acc0 = {}, acc1 = {};
    #pragma unroll
    for (int kk = 0; kk < 5; ++kk) {
        int colbase = tx * 16 + kk * 4 + 2 + hi * 2;  // padded col (= unpadded -2 +PAD_LEFT)
        int r0 = ty0 * 16 + l16;
        int r1 = ty1 * 16 + l16;
        v2f a0, a1;
        a0.x = s_in[r0 * IN_STRIDE + colbase + 0];
        a0.y = s_in[r0 * IN_STRIDE + colbase + 1];
        a1.x = s_in[r1 * IN_STRIDE + colbase + 0];
        a1.y = s_in[r1 * IN_STRIDE + colbase + 1];
        acc0 = __builtin_amdgcn_wmma_f32_16x16x4_f32(false, a0, false, wb[kk],
                                                     (short)0, acc0, false, false);
        acc1 = __builtin_amdgcn_wmma_f32_16x16x4_f32(false, a1, false, wb[kk],
                                                     (short)0, acc1, false, false);
    }
    // C/D layout: VGPR r -> row r (lanes 0-15) / row r+8 (lanes 16-31), col = l16
    #pragma unroll
    for (int r = 0; r < 8; ++r) {
        int rowoff = r + hi * 8 + 2;              // +2 row halo offset
        s_tmp[(ty0 * 16 + rowoff) * TMP_STRIDE + tx * 16 + l16] = acc0[r];
        s_tmp[(ty1 * 16 + rowoff) * TMP_STRIDE + tx * 16 + l16] = acc1[r];
    }
    __syncthreads();

    // ========== vertical pass: Out = Band(16x20) x Tmp(20x16), 2 tiles/wave ==========
    v8f o0 = {}, o1 = {};
    #pragma unroll
    for (int kk = 0; kk < 5; ++kk) {
        int i0 = kk * 4 + hi * 2;                 // padded tmp-row offset within tile
        v2f b0, b1;
        b0.x = s_tmp[(ty0 * 16 + i0    ) * TMP_STRIDE + tx * 16 + l16];
        b0.y = s_tmp[(ty0 * 16 + i0 + 1) * TMP_STRIDE + tx * 16 + l16];
        b1.x = s_tmp[(ty1 * 16 + i0    ) * TMP_STRIDE + tx * 16 + l16];
        b1.y = s_tmp[(ty1 * 16 + i0 + 1) * TMP_STRIDE + tx * 16 + l16];
        o0 = __builtin_amdgcn_wmma_f32_16x16x4_f32(false, wb[kk], false, b0,
                                                   (short)0, o0, false, false);
        o1 = __builtin_amdgcn_wmma_f32_16x16x4_f32(false, wb[kk], false, b1,
                                                   (short)0, o1, false, false);
    }
    // store: per r, lanes 0-15 -> row R cols 0-15, lanes 16-31 -> row R+8 (2x64B segments)
    #pragma unroll
    for (int r = 0; r < 8; ++r) {
        int rowoff = r + hi * 8;
        dst[(ty0 * 16 + rowoff) * W + tx * 16 + l16] = o0[r];
        dst[(ty1 * 16 + rowoff) * W + tx * 16 + l16] = o1[r];
    }
}

extern "C" void kernel_launch(void* const* d_in, const int* in_sizes, int n_in,
                              void* d_out, int out_size, void* d_ws, size_t ws_size,
                              hipStream_t stream) {
    const float* x     = (const float*)d_in[0];   // (16,256,64,64) fp32
    const float* sigma = (const float*)d_in[1];   // scalar
    float* out = (float*)d_out;
    int planes = out_size / (H * W);              // 16*256 = 4096
    gauss_sep_wmma<<<planes, 256, 0, stream>>>(x, sigma, out);
}